// FeedForwardQuantum_65481071395646
// MI455X (gfx1250) — compile-verified
//
#include <hip/hip_runtime.h>

// MI455X / gfx1250, wave32. Fused FFN-with-quantum-activation:
//   proj = x @ w1^T + b1 ; q = cos(proj)*cos(theta) ; out = relu(q @ w2^T + b2)
// Bandwidth-bound (~512 MB traffic, ~22 us floor at 23.3 TB/s). fp32 WMMA
// (V_WMMA_F32_16X16X4_F32) for both skinny GEMMs; one wave owns 16 rows.

typedef float v2f __attribute__((ext_vector_type(2)));
typedef float v8f __attribute__((ext_vector_type(8)));

#define DDIM 1024
#define QDIM 6
#define WAVES_PER_BLOCK 8
#define ROWS_PER_WAVE 16
#define ROWS_PER_BLOCK (WAVES_PER_BLOCK * ROWS_PER_WAVE)

__device__ inline v8f wmma_f32_16x16x4(v2f a, v2f b, v8f c) {
    // 8-arg form: (neg_a, A, neg_b, B, c_mod, C, reuse_a, reuse_b)
    return __builtin_amdgcn_wmma_f32_16x16x4_f32(false, a, false, b, (short)0, c,
                                                 false, false);
}

__global__ __launch_bounds__(256) void ffq_kernel(
    const float* __restrict__ x, const float* __restrict__ w1,
    const float* __restrict__ b1, const float* __restrict__ theta,
    const float* __restrict__ w2, const float* __restrict__ b2,
    float* __restrict__ out) {
    // Per-wave q staging for the GEMM1->GEMM2 lane transpose: [m][k] padded to 16.
    __shared__ float qlds[WAVES_PER_BLOCK][ROWS_PER_WAVE][16];

    const int lane = threadIdx.x & 31;
    const int wave = threadIdx.x >> 5;
    const int half = lane >> 4;   // 0: K lanes {0,1}; 1: K lanes {2,3}
    const int ln   = lane & 15;   // row (A) / column (B,C,D) index within tile
    const int kb   = half * 2;    // this lane's K base inside a 4-wide WMMA step

    const int row0 = blockIdx.x * ROWS_PER_BLOCK + wave * ROWS_PER_WAVE;

    // ---------------- GEMM1: proj[16x16] = x_tile[16x1024] * w1^T (N padded 6->16)
    const float* xrow  = x + (row0 + ln) * DDIM;            // A-matrix row for this lane
    const int    nn    = (ln < QDIM) ? ln : (QDIM - 1);     // clamp to stay in-bounds
    const float* w1row = w1 + nn * DDIM;                    // B column n = w1 row n
    const bool   nvalid = (ln < QDIM);

    v8f acc = {0.f, 0.f, 0.f, 0.f, 0.f, 0.f, 0.f, 0.f};
    for (int k0 = 0; k0 < DDIM; k0 += 16) {
        __builtin_prefetch(xrow + k0 + 128, 0, 0);          // global_prefetch_b8
#pragma unroll
        for (int kk = 0; kk < 16; kk += 4) {
            // A frag: A[m=ln][k0+kk+kb], [..+1]  (8B-aligned float2)
            v2f a = *(const v2f*)(xrow + k0 + kk + kb);
            // B frag: B[k][n] = w1[n*D + k]; zero the padded columns n>=6
            v2f b = *(const v2f*)(w1row + k0 + kk + kb);
            b.x = nvalid ? b.x : 0.f;
            b.y = nvalid ? b.y : 0.f;
            acc = wmma_f32_16x16x4(a, b, acc);
        }
    }

    // ---------------- quantum activation on the C fragment
    // C layout: vgpr j = row (j + 8*half), column ln.  Column index = q index here.
    const float b1v = nvalid ? b1[ln] : 0.f;
    const float ctv = nvalid ? __cosf(theta[ln]) : 0.f;  // cos(theta_q)
#pragma unroll
    for (int j = 0; j < 8; ++j) {
        float qv = __cosf(acc[j] + b1v) * ctv;           // == 0 for padded q >= 6
        qlds[wave][j + 8 * half][ln] = qv;               // transpose via LDS
    }
    __syncthreads();  // uniform across the block; orders DS writes before reads

    // ---------------- GEMM2 A frags (constant across all d-tiles): A[m=ln][k], K padded 6->8
    v2f a21 = *(const v2f*)(&qlds[wave][ln][kb]);        // K 0..3
    v2f a22 = *(const v2f*)(&qlds[wave][ln][4 + kb]);    // K 4..7 (6,7 are zeros)

    // ---------------- GEMM2: out_tile[16x1024] = q[16x8] * w2^T, 64 d-tiles of 16
    for (int d0 = 0; d0 < DDIM; d0 += 16) {
        const int d = d0 + ln;                           // B/C/D column for this lane
        // B frag #1: B[k][n] = w2[d*6 + k], k = kb, kb+1 (all < 6: valid)
        v2f b21;
        b21.x = w2[d * QDIM + kb];
        b21.y = w2[d * QDIM + kb + 1];
        // B frag #2: k = 4+kb, 5+kb; rows 6,7 are zero padding
        v2f b22;
        b22.x = (half == 0) ? w2[d * QDIM + 4] : 0.f;
        b22.y = (half == 0) ? w2[d * QDIM + 5] : 0.f;

        v8f acc2 = {0.f, 0.f, 0.f, 0.f, 0.f, 0.f, 0.f, 0.f};
        acc2 = wmma_f32_16x16x4(a21, b21, acc2);
        acc2 = wmma_f32_16x16x4(a22, b22, acc2);

        const float bias = b2[d];
#pragma unroll
        for (int j = 0; j < 8; ++j) {
            const int m = j + 8 * half;
            float v = acc2[j] + bias;
            v = v > 0.f ? v : 0.f;                       // relu
            out[(row0 + m) * DDIM + d] = v;              // 64B contiguous per half-wave
        }
    }
}

extern "C" void kernel_launch(void* const* d_in, const int* in_sizes, int n_in,
                              void* d_out, int out_size, void* d_ws, size_t ws_size,
                              hipStream_t stream) {
    const float* x     = (const float*)d_in[0];
    const float* w1    = (const float*)d_in[1];
    const float* b1    = (const float*)d_in[2];
    const float* theta = (const float*)d_in[3];
    const float* w2    = (const float*)d_in[4];
    const float* b2    = (const float*)d_in[5];
    float* out = (float*)d_out;

    const long long rows = (long long)in_sizes[0] / DDIM;   // 65536: exact multiple
    const int blocks = (int)(rows / ROWS_PER_BLOCK);        // 512 blocks of 8 waves
    ffq_kernel<<<blocks, WAVES_PER_BLOCK * 32, 0, stream>>>(x, w1, b1, theta, w2,
                                                            b2, out);
}